// GAT_6871947674192
// MI455X (gfx1250) — compile-verified
//
#include <hip/hip_runtime.h>

// GAT forward for MI455X (gfx1250, wave32).
// GEMM uses full-precision V_WMMA_F32_16X16X4_F32 (GEMM is memory-bound at
// ~21 FLOP/B vs 23.3 TB/s HBM, so fp32 WMMA costs nothing vs f16 and keeps
// reference precision). Edge phase uses L2-resident f32 atomics (out = 12.8MB
// << 192MB L2).

#define IN_DIM   128
#define OUT_DIM  64
#define NEG_SLOPE 0.2f

typedef __attribute__((ext_vector_type(2))) float v2f;
typedef __attribute__((ext_vector_type(8))) float v8f;

// Monotone order-preserving float<->int encoding (involution) so segment-max
// can use hardware global_atomic_max_i32 instead of a CAS loop.
__device__ __forceinline__ int enc_f32(float f) {
    int i = __float_as_int(f);
    return i ^ ((i >> 31) & 0x7fffffff);
}
__device__ __forceinline__ float dec_f32(int e) {
    return __int_as_float(e ^ ((e >> 31) & 0x7fffffff));
}

// ---------------------------------------------------------------------------
// Kernel 1: out = bias (broadcast), emax = -inf(enc), denom = 0
// ---------------------------------------------------------------------------
__global__ void gat_init(float* __restrict__ out, const float* __restrict__ bias,
                         int* __restrict__ emax, float* __restrict__ denom,
                         int n_nodes) {
    int i = blockIdx.x * blockDim.x + threadIdx.x;
    int total = n_nodes * OUT_DIM;
    if (i < total) out[i] = bias[i & (OUT_DIM - 1)];
    if (i < n_nodes) { emax[i] = (int)0x80000000; denom[i] = 0.0f; }
}

// ---------------------------------------------------------------------------
// Kernel 2: x = A @ W via fp32 WMMA; fused alpha_src/alpha_dst epilogue.
// One wave computes a 16(M) x 64(N) slab: 4 accumulators of v8f.
// A 16x4 layout: lane l, row = l&15; half h=l>>4 holds K = k+2h, k+2h+1.
// B  4x16 layout: lane l, col = l&15 + 16*t; half h holds same two K rows.
// C/D layout: VGPR j, half h, lane r -> element [m0 + 8h + j][16t + r].
// ---------------------------------------------------------------------------
__global__ __launch_bounds__(256) void gat_gemm_wmma(
    const float* __restrict__ A,       // [N, 128]
    const float* __restrict__ W,       // [128, 64]
    const float* __restrict__ a_src,   // [64]
    const float* __restrict__ a_dst,   // [64]
    float* __restrict__ x,             // [N, 64]
    float* __restrict__ asrc,          // [N]
    float* __restrict__ adst,          // [N]
    int n_nodes, int n_tiles) {
    __shared__ float Wlds[IN_DIM * OUT_DIM];   // 32 KB of 320 KB/WGP
    const int tid = threadIdx.x;
    for (int i = tid; i < IN_DIM * OUT_DIM; i += 256) Wlds[i] = W[i];
    __syncthreads();

    const int wave = tid >> 5;
    const int lane = tid & 31;
    const int r    = lane & 15;
    const int h    = lane >> 4;
    const int tile = blockIdx.x * 8 + wave;
    if (tile >= n_tiles) return;
    const int m0 = tile * 16;

    v8f acc[4] = {};

    const int rowA = min(m0 + r, n_nodes - 1);     // clamp (N=50000 is 16-even)
    const float* __restrict__ arow = A + (size_t)rowA * IN_DIM;

    for (int k = 0; k < IN_DIM; k += 4) {
        const int ka = k + 2 * h;
        v2f a;
        a.x = arow[ka];
        a.y = arow[ka + 1];
#pragma unroll
        for (int t = 0; t < 4; ++t) {
            const int n = t * 16 + r;
            v2f b;
            b.x = Wlds[ka * OUT_DIM + n];
            b.y = Wlds[(ka + 1) * OUT_DIM + n];
            acc[t] = __builtin_amdgcn_wmma_f32_16x16x4_f32(
                /*neg_a=*/false, a, /*neg_b=*/false, b,
                /*c_mod=*/(short)0, acc[t],
                /*reuse_a=*/false, /*reuse_b=*/false);
        }
    }

    // ---- Store x tile --------------------------------------------------
    // Wave-uniform full-tile test: the per-(j,t) offsets are compile-time
    // constants, so the hot path is 32 branch-free global_store_b32 with
    // immediate offsets off one base pointer.
    const bool full = (m0 + 16) <= n_nodes;
    float* __restrict__ xbase = x + (size_t)(m0 + 8 * h) * OUT_DIM + r;
    if (full) {
#pragma unroll
        for (int j = 0; j < 8; ++j)
#pragma unroll
            for (int t = 0; t < 4; ++t)
                xbase[j * OUT_DIM + t * 16] = acc[t][j];
    } else {  // cold: ragged last tile
#pragma unroll
        for (int j = 0; j < 8; ++j)
            if (m0 + 8 * h + j < n_nodes)
#pragma unroll
                for (int t = 0; t < 4; ++t)
                    xbase[j * OUT_DIM + t * 16] = acc[t][j];
    }

    // ---- Fused epilogue: alpha_src[row] = x[row,:] . a_src (same a_dst) --
    float as[4], ad[4];
#pragma unroll
    for (int t = 0; t < 4; ++t) { as[t] = a_src[t * 16 + r]; ad[t] = a_dst[t * 16 + r]; }
    float sj[8], dj[8];
#pragma unroll
    for (int j = 0; j < 8; ++j) {
        float s = 0.0f, d = 0.0f;
#pragma unroll
        for (int t = 0; t < 4; ++t) { s += acc[t][j] * as[t]; d += acc[t][j] * ad[t]; }
        // Reduce across the 16 lanes of each half (halves own different rows).
#pragma unroll
        for (int m = 8; m >= 1; m >>= 1) {
            s += __shfl_xor(s, m, 16);
            d += __shfl_xor(d, m, 16);
        }
        sj[j] = s; dj[j] = d;
    }
    if (r == 0) {
        const int rbase = m0 + 8 * h;
        if (full) {
#pragma unroll
            for (int j = 0; j < 8; ++j) { asrc[rbase + j] = sj[j]; adst[rbase + j] = dj[j]; }
        } else {
#pragma unroll
            for (int j = 0; j < 8; ++j)
                if (rbase + j < n_nodes) { asrc[rbase + j] = sj[j]; adst[rbase + j] = dj[j]; }
        }
    }
}

// ---------------------------------------------------------------------------
// Kernel 3: e = leaky_relu(asrc[src] + adst[dst]); segment max into emax[dst]
// ---------------------------------------------------------------------------
__global__ void gat_edge_max(const int* __restrict__ src, const int* __restrict__ dst,
                             const float* __restrict__ asrc, const float* __restrict__ adst,
                             float* __restrict__ ebuf, int* __restrict__ emax,
                             int n_edges) {
    int e = blockIdx.x * blockDim.x + threadIdx.x;
    if (e >= n_edges) return;
    const int s = src[e], d = dst[e];
    float v = asrc[s] + adst[d];
    v = v > 0.0f ? v : NEG_SLOPE * v;
    ebuf[e] = v;
    atomicMax(&emax[d], enc_f32(v));
}

// ---------------------------------------------------------------------------
// Kernel 4: e_exp = exp(e - emax[dst]); segment sum into denom[dst]
// ---------------------------------------------------------------------------
__global__ void gat_edge_expsum(const int* __restrict__ dst,
                                float* __restrict__ ebuf, const int* __restrict__ emax,
                                float* __restrict__ denom, int n_edges) {
    int e = blockIdx.x * blockDim.x + threadIdx.x;
    if (e >= n_edges) return;
    const int d = dst[e];
    const float ex = expf(ebuf[e] - dec_f32(emax[d]));
    ebuf[e] = ex;
    atomicAdd(&denom[d], ex);
}

// ---------------------------------------------------------------------------
// Kernel 5: out[dst] += (e_exp / denom[dst]) * x[src]; 64 threads per edge
// (coalesced 256B gather per edge; atomics hit the 192MB L2).
// ---------------------------------------------------------------------------
__global__ __launch_bounds__(256) void gat_edge_scatter(
    const int* __restrict__ src, const int* __restrict__ dst,
    const float* __restrict__ x, const float* __restrict__ ebuf,
    const float* __restrict__ denom, float* __restrict__ out, int n_edges) {
    int gid = blockIdx.x * blockDim.x + threadIdx.x;
    int e = gid >> 6;
    int f = gid & 63;
    if (e >= n_edges) return;
    const int s = src[e], d = dst[e];
    const float alpha = ebuf[e] / (denom[d] + 1e-16f);
    atomicAdd(&out[(size_t)d * OUT_DIM + f], alpha * x[(size_t)s * OUT_DIM + f]);
}

// ---------------------------------------------------------------------------
extern "C" void kernel_launch(void* const* d_in, const int* in_sizes, int n_in,
                              void* d_out, int out_size, void* d_ws, size_t ws_size,
                              hipStream_t stream) {
    (void)n_in; (void)out_size; (void)ws_size;
    const float* in_feat = (const float*)d_in[0];
    const int*   edges   = (const int*)d_in[1];   // [2, E]
    const float* W       = (const float*)d_in[2];
    const float* a_src   = (const float*)d_in[3];
    const float* a_dst   = (const float*)d_in[4];
    const float* bias    = (const float*)d_in[5];
    float* out = (float*)d_out;

    const int n_nodes = in_sizes[0] / IN_DIM;
    const int n_edges = in_sizes[1] / 2;
    const int* src = edges;
    const int* dst = edges + n_edges;

    // Workspace layout (~20 MB): x | asrc | adst | emax | denom | ebuf
    float* x     = (float*)d_ws;
    float* asrc  = x + (size_t)n_nodes * OUT_DIM;
    float* adst  = asrc + n_nodes;
    int*   emax  = (int*)(adst + n_nodes);
    float* denom = (float*)(emax + n_nodes);
    float* ebuf  = denom + n_nodes;

    const int n_tiles = (n_nodes + 15) / 16;

    gat_init<<<(n_nodes * OUT_DIM + 255) / 256, 256, 0, stream>>>(
        out, bias, emax, denom, n_nodes);
    gat_gemm_wmma<<<(n_tiles + 7) / 8, 256, 0, stream>>>(
        in_feat, W, a_src, a_dst, x, asrc, adst, n_nodes, n_tiles);
    gat_edge_max<<<(n_edges + 255) / 256, 256, 0, stream>>>(
        src, dst, asrc, adst, ebuf, emax, n_edges);
    gat_edge_expsum<<<(n_edges + 255) / 256, 256, 0, stream>>>(
        dst, ebuf, emax, denom, n_edges);
    gat_edge_scatter<<<(n_edges * 64 + 255) / 256, 256, 0, stream>>>(
        src, dst, x, ebuf, denom, out, n_edges);
}